// DecoderRNN_34333968564390
// MI455X (gfx1250) — compile-verified
//
#include <hip/hip_runtime.h>
#include <math.h>

// Problem dims (match reference)
static constexpr int Bn  = 64;
static constexpr int Pn  = 196;
static constexpr int ENC = 2048;
static constexpr int DEC = 512;
static constexpr int ATT = 512;
static constexpr int EMB = 512;
static constexpr int Vn  = 10000;
static constexpr int Ln  = 30;
static constexpr int Tn  = Ln - 1;          // 29 decode steps
static constexpr int XD  = EMB + ENC;       // 2560 LSTM input width

typedef __attribute__((ext_vector_type(2))) float v2f;
typedef __attribute__((ext_vector_type(8))) float v8f;

#define ACT_NONE 0
#define ACT_SIGMOID 1

// ---------------------------------------------------------------------------
// fp32 WMMA GEMM, N-blocked:  C[M x N] = act( A[M x K] @ W[N x K]^T + bias )
// One wave computes a 16 x 64 tile: 4 accumulators share one A fragment per
// K-step -> 4 v_wmma_f32_16x16x4_f32 per A load (6.4 FLOP/B vs 4 unblocked).
// W row indices of phantom tiles (N not multiple of 64) are clamped (no OOB
// loads); epilogue stores are guarded by n < N. Whole-wave early exit keeps
// EXEC all-ones at every WMMA as the ISA requires.
// A rows may be indirected through arowmap (sorted-batch gather without copy).
// accum!=0 adds into existing C. declen!=nullptr applies per-row time mask.
// M multiple of 16, K multiple of 4 (true for all calls here).
// ---------------------------------------------------------------------------
__global__ void gemm_wmma_f32_kernel(const float* __restrict__ A, int lda,
                                     const int* __restrict__ arowmap,
                                     const float* __restrict__ W, int ldw,
                                     const float* __restrict__ bias,
                                     float* __restrict__ C, int ldc,
                                     int M, int N, int K,
                                     int act, int accum,
                                     const int* __restrict__ declen, int tstep) {
  const int wavesPerBlk = blockDim.x >> 5;
  const int wave = blockIdx.x * wavesPerBlk + (threadIdx.x >> 5);
  const int tilesM  = M >> 4;
  const int tilesN  = (N + 15) >> 4;        // 16-wide N tiles
  const int tilesNG = (tilesN + 3) >> 2;    // groups of 4 (64-wide)
  if (wave >= tilesM * tilesNG) return;     // whole-wave exit -> EXEC all-ones

  const int tm  = wave / tilesNG;
  const int tng = wave % tilesNG;
  const int lane  = threadIdx.x & 31;
  const int lhalf = lane >> 4;              // 0 or 1
  const int lrow  = lane & 15;

  // A fragment (16x4 f32): lane holds row (lane&15), k-pair (lane>>4)*2 + {0,1}
  const int am   = tm * 16 + lrow;
  const int arow = arowmap ? arowmap[am] : am;
  const float* __restrict__ ap = A + (size_t)arow * lda + lhalf * 2;

  // 4 W fragments (4x16 f32 each) from row-major W[N x K]; clamp phantom rows.
  const float* __restrict__ wp0;
  const float* __restrict__ wp1;
  const float* __restrict__ wp2;
  const float* __restrict__ wp3;
  {
    const int nbase = tng * 64 + lrow;
    const int r0 = min(nbase,      N - 1);
    const int r1 = min(nbase + 16, N - 1);
    const int r2 = min(nbase + 32, N - 1);
    const int r3 = min(nbase + 48, N - 1);
    wp0 = W + (size_t)r0 * ldw + lhalf * 2;
    wp1 = W + (size_t)r1 * ldw + lhalf * 2;
    wp2 = W + (size_t)r2 * ldw + lhalf * 2;
    wp3 = W + (size_t)r3 * ldw + lhalf * 2;
  }

  v8f acc0 = {}, acc1 = {}, acc2 = {}, acc3 = {};
#pragma unroll 2
  for (int k = 0; k < K; k += 4) {
    v2f a;  a.x  = ap[k];  a.y  = ap[k + 1];
    v2f b0; b0.x = wp0[k]; b0.y = wp0[k + 1];
    v2f b1; b1.x = wp1[k]; b1.y = wp1[k + 1];
    v2f b2; b2.x = wp2[k]; b2.y = wp2[k + 1];
    v2f b3; b3.x = wp3[k]; b3.y = wp3[k + 1];
    acc0 = __builtin_amdgcn_wmma_f32_16x16x4_f32(false, a, false, b0, (short)0, acc0, false, false);
    acc1 = __builtin_amdgcn_wmma_f32_16x16x4_f32(false, a, false, b1, (short)0, acc1, false, false);
    acc2 = __builtin_amdgcn_wmma_f32_16x16x4_f32(false, a, false, b2, (short)0, acc2, false, false);
    acc3 = __builtin_amdgcn_wmma_f32_16x16x4_f32(false, a, false, b3, (short)0, acc3, false, false);
  }

  // D layout: VGPR r, lanes0-15 -> M=r, lanes16-31 -> M=r+8; N = lane&15
  const float msk = declen ? 0.0f : 1.0f;   // placeholder, per-row below
  (void)msk;
#pragma unroll
  for (int j = 0; j < 4; ++j) {
    const int n = tng * 64 + j * 16 + lrow;
    if (n >= N) continue;                   // store-side guard only
    const v8f acc = (j == 0) ? acc0 : (j == 1) ? acc1 : (j == 2) ? acc2 : acc3;
    const float bv = bias ? bias[n] : 0.0f;
#pragma unroll
    for (int r = 0; r < 8; ++r) {
      const int m = tm * 16 + r + lhalf * 8;
      const size_t off = (size_t)m * ldc + n;
      float v = acc[r] + bv;
      if (accum) v += C[off];
      if (act == ACT_SIGMOID) v = 1.0f / (1.0f + __expf(-v));
      if (declen) v *= (tstep < declen[m]) ? 1.0f : 0.0f;
      C[off] = v;
    }
  }
}

// ---------------------------------------------------------------------------
// Stable argsort by descending length (B=64, one block), emit tail outputs.
// ---------------------------------------------------------------------------
__global__ void sort_kernel(const int* __restrict__ lengths,  // (B,1)
                            const int* __restrict__ caps,     // (B,L)
                            int* __restrict__ sortind, int* __restrict__ declen,
                            float* __restrict__ out_caps,
                            float* __restrict__ out_declen,
                            float* __restrict__ out_sortind) {
  const int i = threadIdx.x;
  if (i < Bn) {
    const int li = lengths[i];
    int rank = 0;
    for (int j = 0; j < Bn; ++j) {
      const int lj = lengths[j];
      if (lj > li || (lj == li && j < i)) ++rank;   // stable descending
    }
    sortind[rank]     = i;
    declen[rank]      = li - 1;
    out_declen[rank]  = (float)(li - 1);
    out_sortind[rank] = (float)i;
  }
  __syncthreads();
  if (i < Bn) {
    const int src = sortind[i];
    for (int t = 0; t < Ln; ++t)
      out_caps[i * Ln + t] = (float)caps[src * Ln + t];
  }
}

__global__ void rowmap_kernel(const int* __restrict__ sortind, int* __restrict__ rowmap) {
  const int id = blockIdx.x * blockDim.x + threadIdx.x;
  if (id < Bn * Pn) rowmap[id] = sortind[id / Pn] * Pn + (id % Pn);
}

// mean over P of sorted encoder rows -> (B, ENC)
__global__ void mean_kernel(const float* __restrict__ enc,
                            const int* __restrict__ sortind,
                            float* __restrict__ meanout) {
  const int id = blockIdx.x * blockDim.x + threadIdx.x;  // B*ENC threads
  const int b = id / ENC, e = id % ENC;
  const float* ep = enc + ((size_t)sortind[b] * Pn) * ENC + e;
  float s = 0.0f;
  for (int p = 0; p < Pn; ++p) s += ep[(size_t)p * ENC];
  meanout[id] = s * (1.0f / (float)Pn);
}

// ---------------------------------------------------------------------------
// e[b,p] = relu(att1[b,p,:] + att2[b,:]) . w_full + b_full ; softmax over p.
// Writes full alpha to ws (for awe) and masked alpha to d_out.
// One block (8 waves) per batch row.
// ---------------------------------------------------------------------------
__global__ void attn_kernel(const float* __restrict__ att1,
                            const float* __restrict__ att2,
                            const float* __restrict__ w_full,
                            const float* __restrict__ b_full,
                            float* __restrict__ alpha_ws,
                            float* __restrict__ alphas_out,
                            const int* __restrict__ declen, int tstep) {
  __shared__ float e[Pn];
  __shared__ float red[2];
  const int b    = blockIdx.x;
  const int lane = threadIdx.x & 31;
  const int wave = threadIdx.x >> 5;

  const float* __restrict__ a2 = att2 + (size_t)b * ATT;
  for (int p = wave; p < Pn; p += 8) {
    const float* __restrict__ a1 = att1 + ((size_t)b * Pn + p) * ATT;
    float s = 0.0f;
    for (int c = lane; c < ATT; c += 32) {
      float v = a1[c] + a2[c];
      v = v > 0.0f ? v : 0.0f;
      s += v * w_full[c];
    }
    for (int off = 16; off > 0; off >>= 1) s += __shfl_xor(s, off, 32);
    if (lane == 0) e[p] = s + b_full[0];
  }
  __syncthreads();

  if (threadIdx.x < 32) {
    float m = -1e30f;
    for (int p = lane; p < Pn; p += 32) m = fmaxf(m, e[p]);
    for (int off = 16; off > 0; off >>= 1) m = fmaxf(m, __shfl_xor(m, off, 32));
    if (lane == 0) red[0] = m;
  }
  __syncthreads();
  const float mx = red[0];
  for (int p = threadIdx.x; p < Pn; p += blockDim.x) e[p] = __expf(e[p] - mx);
  __syncthreads();
  if (threadIdx.x < 32) {
    float s = 0.0f;
    for (int p = lane; p < Pn; p += 32) s += e[p];
    for (int off = 16; off > 0; off >>= 1) s += __shfl_xor(s, off, 32);
    if (lane == 0) red[1] = 1.0f / s;
  }
  __syncthreads();
  const float inv = red[1];
  const float msk = (tstep < declen[b]) ? 1.0f : 0.0f;
  for (int p = threadIdx.x; p < Pn; p += blockDim.x) {
    const float a = e[p] * inv;
    alpha_ws[b * Pn + p] = a;
    alphas_out[((size_t)b * Tn + tstep) * Pn + p] = a * msk;
  }
}

// Build LSTM input x = [ emb(cap_t) | gate * (alpha . enc) ]  -> (B, 2560)
__global__ void xbuild_kernel(const float* __restrict__ enc,
                              const float* __restrict__ emb_table,
                              const int* __restrict__ caps,
                              const int* __restrict__ sortind,
                              const float* __restrict__ alpha,
                              const float* __restrict__ gate,
                              float* __restrict__ x, int tstep) {
  const int id = blockIdx.x * blockDim.x + threadIdx.x;  // B*XD threads
  const int b = id / XD, j = id % XD;
  const int bs = sortind[b];
  if (j < EMB) {
    const int cap = caps[bs * Ln + tstep];
    x[id] = emb_table[(size_t)cap * EMB + j];
  } else {
    const int e = j - EMB;
    const float* __restrict__ ep = enc + ((size_t)bs * Pn) * ENC + e;
    const float* __restrict__ ap = alpha + b * Pn;
    float s = 0.0f;
    for (int p = 0; p < Pn; ++p) s += ap[p] * ep[(size_t)p * ENC];
    x[id] = gate[b * ENC + e] * s;
  }
}

__device__ __forceinline__ float sigf(float v) { return 1.0f / (1.0f + __expf(-v)); }

// LSTM pointwise update (gates layout: [i | f | g | o], each DEC wide)
__global__ void lstm_kernel(const float* __restrict__ gates,
                            float* __restrict__ h, float* __restrict__ c) {
  const int id = blockIdx.x * blockDim.x + threadIdx.x;  // B*DEC threads
  const int b = id / DEC, j = id % DEC;
  const float* __restrict__ g = gates + (size_t)b * 4 * DEC;
  const float i  = sigf(g[j]);
  const float f  = sigf(g[DEC + j]);
  const float gg = tanhf(g[2 * DEC + j]);
  const float o  = sigf(g[3 * DEC + j]);
  const float cn = f * c[id] + i * gg;
  c[id] = cn;
  h[id] = o * tanhf(cn);
}

// ---------------------------------------------------------------------------
// Host orchestration
// ---------------------------------------------------------------------------
static inline void launch_gemm(hipStream_t s, const float* A, int lda, const int* rmap,
                               const float* W, int ldw, const float* bias,
                               float* C, int ldc, int M, int N, int K,
                               int act = ACT_NONE, int accum = 0,
                               const int* declen = nullptr, int tstep = 0) {
  const int tilesM  = M / 16;
  const int tilesN  = (N + 15) / 16;
  const int tilesNG = (tilesN + 3) / 4;     // 64-wide N groups per wave
  const int tiles   = tilesM * tilesNG;
  const int blocks  = (tiles + 7) / 8;      // 8 waves (256 threads) per block
  gemm_wmma_f32_kernel<<<blocks, 256, 0, s>>>(A, lda, rmap, W, ldw, bias,
                                              C, ldc, M, N, K, act, accum,
                                              declen, tstep);
}

extern "C" void kernel_launch(void* const* d_in, const int* in_sizes, int n_in,
                              void* d_out, int out_size, void* d_ws, size_t ws_size,
                              hipStream_t stream) {
  (void)in_sizes; (void)n_in; (void)out_size; (void)ws_size;

  const float* enc      = (const float*)d_in[0];
  const int*   caps     = (const int*)  d_in[1];
  const int*   lens     = (const int*)  d_in[2];
  const float* W_enc    = (const float*)d_in[3];
  const float* b_enc    = (const float*)d_in[4];
  const float* W_dec    = (const float*)d_in[5];
  const float* b_dec    = (const float*)d_in[6];
  const float* w_full   = (const float*)d_in[7];
  const float* b_full   = (const float*)d_in[8];
  const float* emb_tab  = (const float*)d_in[9];
  const float* W_ih     = (const float*)d_in[10];
  const float* b_ih     = (const float*)d_in[11];
  const float* W_hh     = (const float*)d_in[12];
  const float* b_hh     = (const float*)d_in[13];
  const float* W_init_h = (const float*)d_in[14];
  const float* b_init_h = (const float*)d_in[15];
  const float* W_init_c = (const float*)d_in[16];
  const float* b_init_c = (const float*)d_in[17];
  const float* W_beta   = (const float*)d_in[18];
  const float* b_beta   = (const float*)d_in[19];
  const float* W_fc     = (const float*)d_in[20];
  const float* b_fc     = (const float*)d_in[21];

  // Output layout: preds (B,T,V) | alphas (B,T,P) | caps (B,L) | declen (B) | sortind (B)
  float* preds_out   = (float*)d_out;
  float* alphas_out  = preds_out  + (size_t)Bn * Tn * Vn;
  float* caps_out    = alphas_out + (size_t)Bn * Tn * Pn;
  float* declen_out  = caps_out   + (size_t)Bn * Ln;
  float* sortind_out = declen_out + Bn;

  // Workspace carve (floats then ints); ~28 MB total
  float* w = (float*)d_ws;
  float* att1  = w; w += (size_t)Bn * Pn * ATT;   // 12544 x 512
  float* meanE = w; w += (size_t)Bn * ENC;
  float* h     = w; w += (size_t)Bn * DEC;
  float* c     = w; w += (size_t)Bn * DEC;
  float* att2  = w; w += (size_t)Bn * ATT;
  float* gate  = w; w += (size_t)Bn * ENC;
  float* alpha = w; w += (size_t)Bn * Pn;
  float* x     = w; w += (size_t)Bn * XD;
  float* gates = w; w += (size_t)Bn * 4 * DEC;
  int* iw      = (int*)w;
  int* sortind = iw; iw += Bn;
  int* declen  = iw; iw += Bn;
  int* rowmap  = iw; iw += Bn * Pn;

  // --- setup ---
  sort_kernel<<<1, Bn, 0, stream>>>(lens, caps, sortind, declen,
                                    caps_out, declen_out, sortind_out);
  rowmap_kernel<<<(Bn * Pn + 255) / 256, 256, 0, stream>>>(sortind, rowmap);
  mean_kernel<<<(Bn * ENC) / 256, 256, 0, stream>>>(enc, sortind, meanE);

  // att1 = enc_sorted @ W_enc^T + b_enc   (12544 x 512, K=2048)
  launch_gemm(stream, enc, ENC, rowmap, W_enc, ENC, b_enc,
              att1, ATT, Bn * Pn, ATT, ENC);
  // h0, c0 from mean encoder output (64 x 512, K=2048)
  launch_gemm(stream, meanE, ENC, nullptr, W_init_h, ENC, b_init_h,
              h, DEC, Bn, DEC, ENC);
  launch_gemm(stream, meanE, ENC, nullptr, W_init_c, ENC, b_init_c,
              c, DEC, Bn, DEC, ENC);

  // --- recurrence ---
  for (int t = 0; t < Tn; ++t) {
    // att2 = h @ W_dec^T + b_dec        (64 x 512, K=512)
    launch_gemm(stream, h, DEC, nullptr, W_dec, DEC, b_dec,
                att2, ATT, Bn, ATT, DEC);
    // gate = sigmoid(h @ W_beta^T + b_beta)  (64 x 2048, K=512)
    launch_gemm(stream, h, DEC, nullptr, W_beta, DEC, b_beta,
                gate, ENC, Bn, ENC, DEC, ACT_SIGMOID);
    // attention scores + softmax (alpha), masked alpha -> d_out
    attn_kernel<<<Bn, 256, 0, stream>>>(att1, att2, w_full, b_full,
                                        alpha, alphas_out, declen, t);
    // x = [ emb | gate * (alpha . enc) ]
    xbuild_kernel<<<(Bn * XD) / 256, 256, 0, stream>>>(enc, emb_tab, caps, sortind,
                                                       alpha, gate, x, t);
    // gates = x @ W_ih^T + b_ih         (64 x 2048, K=2560)
    launch_gemm(stream, x, XD, nullptr, W_ih, XD, b_ih,
                gates, 4 * DEC, Bn, 4 * DEC, XD);
    // gates += h @ W_hh^T + b_hh        (64 x 2048, K=512)
    launch_gemm(stream, h, DEC, nullptr, W_hh, DEC, b_hh,
                gates, 4 * DEC, Bn, 4 * DEC, DEC, ACT_NONE, /*accum=*/1);
    // LSTM pointwise: update h, c
    lstm_kernel<<<(Bn * DEC) / 256, 256, 0, stream>>>(gates, h, c);
    // preds = (h_new @ W_fc^T + b_fc) * mask_t -> d_out  (64 x 10000, K=512)
    launch_gemm(stream, h, DEC, nullptr, W_fc, DEC, b_fc,
                preds_out + (size_t)t * Vn, Tn * Vn, Bn, Vn, DEC,
                ACT_NONE, 0, declen, t);
  }
}